// Net_81939386073093
// MI455X (gfx1250) — compile-verified
//
#include <hip/hip_runtime.h>
#include <hip/hip_bf16.h>

typedef __attribute__((ext_vector_type(2))) float v2f;
typedef __attribute__((ext_vector_type(8))) float v8f;

__device__ __forceinline__ float sigmoidf(float v) {
    return 1.0f / (1.0f + __expf(-v));
}

// Split-K dual matvec on V_WMMA_F32_16X16X4_F32, software-pipelined.
// For a 16-row tile of W [fout x fin] and K-chunk [s*chunk, (s+1)*chunk):
//   pa [s][rows] = partial of W @ u
//   pda[s][rows] = partial of W @ v
// Disjoint stores per (tile, slice) -> deterministic (no atomics).
//
// WMMA operand mapping (wave32):
//   A (16x4 f32): lane%16 = M, lane/16 selects K pair {0,1}/{2,3} -> one b64/lane
//   B (4x16 f32): lane%16 = N; column N=0 carries u, N=1 carries v, rest zero
//   C (16x16 f32): VGPR r, lanes 0-15: M=r, lanes 16-31: M=8+r, N=lane&15
__global__ __launch_bounds__(256) void dual_matvec_wmma_part(
    const float* __restrict__ W,
    const float* __restrict__ u, const float* __restrict__ v,
    float* __restrict__ pa, float* __restrict__ pda,
    int fout, int fin, int chunk)
{
    const int lane = threadIdx.x & 31;
    const int wid  = threadIdx.x >> 5;
    const int tile = blockIdx.x * (blockDim.x >> 5) + wid;
    if (tile * 16 >= fout) return;          // wave-uniform exit

    const int s    = blockIdx.y;            // K-slice index
    const int kbeg = s * chunk;
    const int m = tile * 16;
    const int n = lane & 15;                // B column / A row selector
    const int h = lane >> 4;                // lane half: K pair {0,1} vs {2,3}

    // Loop-invariant per-lane base pointers -> loads become base+imm offsets.
    const float* wrow  = W + (size_t)(m + n) * fin + (size_t)kbeg + 2 * h;
    const float* bsrc  = (n == 1) ? v : u;  // branch-free operand select
    const float* bbase = bsrc + (size_t)kbeg + 2 * h;
    const bool   live  = (n < 2);

    v8f acc = {0.f, 0.f, 0.f, 0.f, 0.f, 0.f, 0.f, 0.f};

    // Software pipeline: group = 16 K-steps = 4 WMMAs = 8 b64 loads.
    // Current group lives in ca/cb while the next group's loads are in flight,
    // so each WMMA only waits on loads issued one full iteration earlier.
    v2f ca[4], cb[4];
#pragma unroll
    for (int i = 0; i < 4; ++i) {
        ca[i] = __builtin_nontemporal_load((const v2f*)(wrow  + 4 * i));
        cb[i] = *(const v2f*)(bbase + 4 * i);
    }

    for (int k0 = 0; k0 < chunk; k0 += 16) {
        const int kn = (k0 + 16 < chunk) ? (k0 + 16) : k0;  // next group (clamped)
        // stream-ahead prefetch of the weight row (speculative, OOB-safe)
        __builtin_prefetch(wrow + k0 + 4096, 0, 3);

        v2f na[4], nb[4];
#pragma unroll
        for (int i = 0; i < 4; ++i) {
            // A: weights consumed exactly once -> non-temporal stream
            na[i] = __builtin_nontemporal_load((const v2f*)(wrow  + kn + 4 * i));
            // B: hot operand vectors, default temporal policy, aligned b64
            nb[i] = *(const v2f*)(bbase + kn + 4 * i);
        }

#pragma unroll
        for (int i = 0; i < 4; ++i) {
            v2f b;
            b.x = live ? cb[i].x : 0.0f;                // mask via cndmask,
            b.y = live ? cb[i].y : 0.0f;                // EXEC stays all-ones
            acc = __builtin_amdgcn_wmma_f32_16x16x4_f32(
                false, ca[i], false, b, (short)0, acc, false, false);
        }

#pragma unroll
        for (int i = 0; i < 4; ++i) { ca[i] = na[i]; cb[i] = nb[i]; }
    }

    // C: lane n==0 holds partial W@u, lane n==1 holds partial W@v
    const int rbase = m + h * 8;
#pragma unroll
    for (int r = 0; r < 8; ++r) {
        const size_t off = (size_t)s * fout + (rbase + r);
        if (n == 0) pa[off]  = acc[r];
        if (n == 1) pda[off] = acc[r];
    }
}

// Sums K-slice partials in fixed order (deterministic), adds bias, applies
// the fused sigmoid-JVP epilogue (or linear for the last layer).
__global__ __launch_bounds__(256) void combine_bias_act(
    const float* __restrict__ pa, const float* __restrict__ pda,
    const float* __restrict__ bias,
    float* __restrict__ y, float* __restrict__ dy,
    int fout, int S, int fuse_sigmoid)
{
    const int row = blockIdx.x * 256 + threadIdx.x;
    if (row >= fout) return;
    float a = 0.0f, da = 0.0f;
    for (int s = 0; s < S; ++s) {
        a  += pa[(size_t)s * fout + row];
        da += pda[(size_t)s * fout + row];
    }
    a += bias[row];
    if (fuse_sigmoid) {
        float sg = sigmoidf(a);
        y[row]  = sg;
        dy[row] = sg * (1.0f - sg) * da;
    } else {
        y[row]  = a;
        dy[row] = da;
    }
}

// theta = [1(x16), z, z_i z_j (i<=j), z_i z_j z_k (i<=j<=k)] (984 terms),
// dzb = E_w @ theta + E_b.  Also emits z, dz, dzb into d_out[0..47].
__global__ __launch_bounds__(256) void sindy_dzb_kernel(
    const float* __restrict__ z_in, const float* __restrict__ dz_in,
    const float* __restrict__ E_w, const float* __restrict__ E_b,
    float* __restrict__ out, float* __restrict__ dzb_ws)
{
    __shared__ float theta[984];
    __shared__ float zsh[16];
    __shared__ float part[16][17];

    const int t = threadIdx.x;
    if (t < 16) zsh[t] = z_in[t];
    __syncthreads();

    for (int idx = t; idx < 984; idx += 256) {
        float val;
        if (idx < 16) {
            val = 1.0f;
        } else if (idx < 32) {
            val = zsh[idx - 16];
        } else if (idx < 168) {           // 136 upper-tri pairs
            int p = idx - 32, i = 0;
            while (p >= 16 - i) { p -= 16 - i; ++i; }
            val = zsh[i] * zsh[i + p];
        } else {                          // 816 upper-tri triples
            int p = idx - 168, i = 0;
            for (;;) { int nn = 16 - i; int c = nn * (nn + 1) / 2;
                       if (p < c) break; p -= c; ++i; }
            int j = i;
            for (;;) { int c = 16 - j; if (p < c) break; p -= c; ++j; }
            val = zsh[i] * zsh[j] * zsh[j + p];
        }
        theta[idx] = val;
    }
    __syncthreads();

    const int o = t & 15;
    const int c = t >> 4;
    float acc = 0.0f;
    const float* row = E_w + (size_t)o * 984;
    for (int q = c; q < 984; q += 16) acc += row[q] * theta[q];
    part[c][o] = acc;
    __syncthreads();

    if (t < 16) {
        float s = E_b[t];
#pragma unroll
        for (int cc = 0; cc < 16; ++cc) s += part[cc][t];
        dzb_ws[t]   = s;
        out[32 + t] = s;        // dzb
        out[t]      = z_in[t];  // z
        out[16 + t] = dz_in[t]; // dz
    }
}

static inline void run_layer(const float* W, const float* b,
                             const float* u, const float* v,
                             float* y, float* dy,
                             int fout, int fin, int S, int fuse,
                             float* pa, float* pda, hipStream_t stream)
{
    const int tiles = fout / 16;
    const int wpb   = (tiles >= 8) ? 8 : tiles;       // waves per block
    dim3 blk(wpb * 32);
    dim3 grd((tiles + wpb - 1) / wpb, S);
    dual_matvec_wmma_part<<<grd, blk, 0, stream>>>(W, u, v, pa, pda,
                                                   fout, fin, fin / S);
    combine_bias_act<<<(fout + 255) / 256, 256, 0, stream>>>(
        pa, pda, b, y, dy, fout, S, fuse);
}

extern "C" void kernel_launch(void* const* d_in, const int* in_sizes, int n_in,
                              void* d_out, int out_size, void* d_ws, size_t ws_size,
                              hipStream_t stream) {
    const float* x     = (const float*)d_in[0];
    const float* dx    = (const float*)d_in[1];
    /* d_in[2] = ddx : unused by the reference outputs */
    const float* we_w0 = (const float*)d_in[3];
    const float* we_b0 = (const float*)d_in[4];
    const float* we_w1 = (const float*)d_in[5];
    const float* we_b1 = (const float*)d_in[6];
    const float* we_w2 = (const float*)d_in[7];
    const float* we_b2 = (const float*)d_in[8];
    const float* wd_w0 = (const float*)d_in[9];
    const float* wd_b0 = (const float*)d_in[10];
    const float* wd_w1 = (const float*)d_in[11];
    const float* wd_b1 = (const float*)d_in[12];
    const float* wd_w2 = (const float*)d_in[13];
    const float* wd_b2 = (const float*)d_in[14];
    const float* E_w   = (const float*)d_in[15];
    const float* E_b   = (const float*)d_in[16];

    float* out = (float*)d_out;
    float* ws  = (float*)d_ws;

    // workspace layout (floats)
    float* h1  = ws;            // 4096
    float* dh1 = ws + 4096;     // 4096
    float* h2  = ws + 8192;     // 1024
    float* dh2 = ws + 9216;     // 1024
    float* z   = ws + 10240;    // 16
    float* dz  = ws + 10256;    // 16
    float* dzb = ws + 10272;    // 16
    float* g1  = ws + 10288;    // 1024
    float* dg1 = ws + 11312;    // 1024
    float* g2  = ws + 12336;    // 4096
    float* dg2 = ws + 16432;    // 4096
    float* pa  = ws + 20544;    // up to 65536 (fout*S partials, u-path)
    float* pda = ws + 20544 + 65536;  // up to 65536 (v-path)

    // Split-K needs room for the partial buffers; else run single-slice.
    const bool big = ws_size >= (size_t)(20544 + 2 * 65536) * sizeof(float);
    const int Se0 = big ? 8 : 1;    // 4096x16384 -> 2048 waves
    const int Se1 = big ? 16 : 1;   // 1024x4096  -> 1024 waves
    const int Se2 = big ? 8 : 1;    // 16x1024
    const int Sd0 = 1;              // 1024x16 (K too small to split)
    const int Sd1 = big ? 8 : 1;    // 4096x1024 -> 2048 waves
    const int Sd2 = big ? 4 : 1;    // 16384x4096 -> 4096 waves

    // Encoder JVP chain: (x, dx) -> (h1, dh1) -> (h2, dh2) -> (z, dz)
    run_layer(we_w0, we_b0, x,  dx,  h1, dh1, 4096, 16384, Se0, 1, pa, pda, stream);
    run_layer(we_w1, we_b1, h1, dh1, h2, dh2, 1024,  4096, Se1, 1, pa, pda, stream);
    run_layer(we_w2, we_b2, h2, dh2, z,  dz,    16,  1024, Se2, 0, pa, pda, stream);

    // SINDy library + regression: dzb = E_w @ theta(z) + E_b ; emit z,dz,dzb
    sindy_dzb_kernel<<<1, 256, 0, stream>>>(z, dz, E_w, E_b, out, dzb);

    // Decoder JVP chain: (z, dzb) -> (g1, dg1) -> (g2, dg2) -> (xb, dxb)
    run_layer(wd_w0, wd_b0, z,  dzb, g1, dg1, 1024,    16, Sd0, 1, pa, pda, stream);
    run_layer(wd_w1, wd_b1, g1, dg1, g2, dg2, 4096,  1024, Sd1, 1, pa, pda, stream);
    run_layer(wd_w2, wd_b2, g2, dg2, out + 48, out + 16432,
              16384, 4096, Sd2, 0, pa, pda, stream);
}